// VarianceExpandingDiffusionSUN_35330400977186
// MI455X (gfx1250) — compile-verified
//
#include <hip/hip_runtime.h>
#include <stdint.h>

// ----------------------------------------------------------------------------
// VarianceExpandingDiffusionSUN: Ut = V diag(exp(i*xs)) V^H U0, complex 3x3,
// B = 524288 batches. Memory-bound (~120 MB traffic @ 23.3 TB/s => ~5.2 us
// floor, only ~262 MFLOP). Strategy: stream the four matrix arrays through a
// double-buffered async global->LDS pipeline (CDNA5 GLOBAL_LOAD_ASYNC_TO_LDS,
// ASYNCcnt), compute the tiny complex matmuls fully unrolled in registers.
// ----------------------------------------------------------------------------

#define NC 3
#define TILE_B 256                           // batches per tile == block size
#define TPB 4                                // tiles per block (pipeline depth amortization)
#define MAT_BYTES (TILE_B * NC * NC * 4)     // 9216 B per matrix-array tile
#define CHUNKS_PER_MAT (MAT_BYTES / 16)      // 576 x b128 chunks
#define TOTAL_CHUNKS (4 * CHUNKS_PER_MAT)    // 2304
#define CH_PER_THR (TOTAL_CHUNKS / TILE_B)   // 9 async b128 per thread per tile

typedef int v4i_t __attribute__((ext_vector_type(4)));
typedef __attribute__((address_space(1))) v4i_t* gv4i_ptr;
typedef __attribute__((address_space(3))) v4i_t* lv4i_ptr;

// ---- CDNA5 async copy global -> LDS (ASYNCcnt-tracked) ---------------------
#if defined(__HIP_DEVICE_COMPILE__) && __has_builtin(__builtin_amdgcn_global_load_async_to_lds_b128)
#define ASYNC_CP_B128(gsrc, ldst)                                               \
  __builtin_amdgcn_global_load_async_to_lds_b128(                               \
      (gv4i_ptr)(uintptr_t)(gsrc),                                              \
      (lv4i_ptr)(uint32_t)(uintptr_t)(ldst),                                    \
      0, 0)
#else
// VDST = LDS byte address (low 32 bits of generic LDS pointer), VADDR = 64-bit
// global address, SADDR = off.  Mnemonic per CDNA5 ISA 15.18.3 (opcode 98).
#define ASYNC_CP_B128(gsrc, ldst)                                               \
  asm volatile("global_load_async_to_lds_b128 %0, %1, off"                      \
               :: "v"((uint32_t)(uintptr_t)(ldst)),                             \
                  "v"((const void*)(gsrc))                                      \
               : "memory")
#endif

#if defined(__HIP_DEVICE_COMPILE__) && __has_builtin(__builtin_amdgcn_s_wait_asynccnt)
#define WAIT_ASYNC(n) __builtin_amdgcn_s_wait_asynccnt(n)
#else
#define WAIT_ASYNC(n) asm volatile("s_wait_asynccnt %0" :: "i"(n) : "memory")
#endif

__global__ __launch_bounds__(TILE_B)
void VarianceExpandingDiffusionSUN_kernel(const float* __restrict__ U0re,
                                          const float* __restrict__ U0im,
                                          const float* __restrict__ xs,
                                          const float* __restrict__ Vre,
                                          const float* __restrict__ Vim,
                                          float* __restrict__ out,
                                          int ntiles_total) {
  // double-buffered staging for {U0re, U0im, Vre, Vim} tiles: 2 * 36 KB
  __shared__ __align__(16) char smem[2 * 4 * MAT_BYTES];

  const int tid = threadIdx.x;
  int tile0 = blockIdx.x * TPB;
  int nt = ntiles_total - tile0;
  if (nt <= 0) return;
  if (nt > TPB) nt = TPB;

  // ---- tile-invariant per-thread async chunk descriptors -------------------
  // Flat chunk id c in [0, 2304): matrix m = c/576, within-matrix chunk w.
  // Every thread owns exactly 9 chunks => every wave issues exactly 9 async
  // ops per tile (uniform ASYNCcnt accounting).
  const char* gbase[CH_PER_THR];
  uint32_t    loff[CH_PER_THR];
#pragma unroll
  for (int j = 0; j < CH_PER_THR; ++j) {
    int c = tid + TILE_B * j;
    int m = c / CHUNKS_PER_MAT;
    int w = c - m * CHUNKS_PER_MAT;
    const float* bp = (m == 0) ? U0re : (m == 1) ? U0im : (m == 2) ? Vre : Vim;
    gbase[j] = (const char*)bp + (size_t)w * 16;
    loff[j]  = (uint32_t)(m * MAT_BYTES + w * 16);
  }

  auto issue_tile = [&](int tile, int buf) {
    const size_t goff = (size_t)tile * MAT_BYTES;
    char* lbuf = smem + (size_t)buf * (4 * MAT_BYTES);
#pragma unroll
    for (int j = 0; j < CH_PER_THR; ++j) {
      ASYNC_CP_B128(gbase[j] + goff, lbuf + loff[j]);
    }
  };

  issue_tile(tile0, 0);  // prologue: prefetch first tile

  for (int it = 0; it < nt; ++it) {
    // software pipeline: launch next tile's async copies, then wait until only
    // those remain outstanding (async loads complete in order per wave).
    if (it + 1 < nt) {
      issue_tile(tile0 + it + 1, (it + 1) & 1);
      WAIT_ASYNC(CH_PER_THR);
    } else {
      WAIT_ASYNC(0);
    }
    __syncthreads();  // LDS tile written cooperatively by all 8 waves

    const float* lb   = (const float*)(smem + (size_t)(it & 1) * (4 * MAT_BYTES));
    const float* lu0r = lb;
    const float* lu0i = lb + 1 * TILE_B * 9;
    const float* lvr  = lb + 2 * TILE_B * 9;
    const float* lvi  = lb + 3 * TILE_B * 9;

    const size_t b = (size_t)(tile0 + it) * TILE_B + tid;
    const int base9 = tid * 9;

    // registers: stride-9-dword LDS reads are bank-conflict-free (gcd(9,64)=1)
    float vr[9], vi[9], ur[9], ui[9];
#pragma unroll
    for (int k = 0; k < 9; ++k) {
      ur[k] = lu0r[base9 + k];
      ui[k] = lu0i[base9 + k];
      vr[k] = lvr[base9 + k];
      vi[k] = lvi[base9 + k];
    }

    // phases = exp(i*xs)
    float ps[3], pc[3];
#pragma unroll
    for (int k = 0; k < 3; ++k) {
      float x = xs[b * 3 + k];
      __sincosf(x, &ps[k], &pc[k]);
    }

    // W = V * diag(phases)
    float wr[9], wi[9];
#pragma unroll
    for (int i = 0; i < 3; ++i)
#pragma unroll
      for (int k = 0; k < 3; ++k) {
        const float a = vr[i * 3 + k], bb = vi[i * 3 + k];
        wr[i * 3 + k] = a * pc[k] - bb * ps[k];
        wi[i * 3 + k] = a * ps[k] + bb * pc[k];
      }

    // E = W @ V^H :  E[i][j] = sum_k W[i][k] * conj(V[j][k])
    float er[9], ei[9];
#pragma unroll
    for (int i = 0; i < 3; ++i)
#pragma unroll
      for (int j = 0; j < 3; ++j) {
        float re = 0.f, im = 0.f;
#pragma unroll
        for (int k = 0; k < 3; ++k) {
          re += wr[i * 3 + k] * vr[j * 3 + k] + wi[i * 3 + k] * vi[j * 3 + k];
          im += wi[i * 3 + k] * vr[j * 3 + k] - wr[i * 3 + k] * vi[j * 3 + k];
        }
        er[i * 3 + j] = re;
        ei[i * 3 + j] = im;
      }

    // Ut = E @ U0, stored interleaved [re, im] -> contiguous 72 B per thread
    float2* o = ((float2*)out) + b * 9;
#pragma unroll
    for (int i = 0; i < 3; ++i)
#pragma unroll
      for (int j = 0; j < 3; ++j) {
        float re = 0.f, im = 0.f;
#pragma unroll
        for (int k = 0; k < 3; ++k) {
          re += er[i * 3 + k] * ur[k * 3 + j] - ei[i * 3 + k] * ui[k * 3 + j];
          im += er[i * 3 + k] * ui[k * 3 + j] + ei[i * 3 + k] * ur[k * 3 + j];
        }
        o[i * 3 + j] = make_float2(re, im);
      }

    __syncthreads();  // protect LDS buffer reuse before next iteration's issue
  }
}

extern "C" void kernel_launch(void* const* d_in, const int* in_sizes, int n_in,
                              void* d_out, int out_size, void* d_ws, size_t ws_size,
                              hipStream_t stream) {
  (void)n_in; (void)out_size; (void)d_ws; (void)ws_size;
  const float* U0re = (const float*)d_in[0];
  const float* U0im = (const float*)d_in[1];
  const float* xs   = (const float*)d_in[2];
  const float* Vre  = (const float*)d_in[3];
  const float* Vim  = (const float*)d_in[4];
  float* out = (float*)d_out;

  const int nb = in_sizes[2] / NC;        // B = 524288
  const int ntiles = nb / TILE_B;         // 2048 (B is a multiple of 256)
  const int grid = (ntiles + TPB - 1) / TPB;  // 512 blocks x 8 waves

  hipLaunchKernelGGL(VarianceExpandingDiffusionSUN_kernel,
                     dim3(grid), dim3(TILE_B), 0, stream,
                     U0re, U0im, xs, Vre, Vim, out, ntiles);
}